// FFMEncodingLayer_75909251989907
// MI455X (gfx1250) — compile-verified
//
#include <hip/hip_runtime.h>

typedef float v2f __attribute__((ext_vector_type(2)));
typedef float v8f __attribute__((ext_vector_type(8)));

#define N_NODES 50000
#define NTILES  3125          // 50000 / 16 exactly
#define F_OUT   96
#define K_PROJ  256           // 2 * mapping_size
#define K_FUS   192           // 2 * F_OUT
#define BN_EPS  1e-5f

// ---------------------------------------------------------------------------
// Kernel 1: Fourier feature mapping + projection GEMM.
// One wave (32 threads) per 16-node row tile. A (16x256 ffm tile) staged in
// LDS, B (W_proj) read through L1/L2 (96 KB, fully resident). fp32 WMMA
// 16x16x4, 6 column tiles of 16 -> 96 output features.
// ---------------------------------------------------------------------------
__global__ __launch_bounds__(32) void ffm_proj_kernel(
    const float* __restrict__ x, const float* __restrict__ B,
    const float* __restrict__ Wp, const float* __restrict__ bp,
    float* __restrict__ x_embed)
{
    __shared__ float ffm[16 * K_PROJ];
    const int lane = threadIdx.x;
    const int row0 = blockIdx.x * 16;

    // Build the 16x256 [cos | sin] tile. u = x @ B ; hw sin/cos take input in
    // revolutions, so sinf_hw(u) == sin(2*pi*u) with no explicit 2*pi multiply.
    for (int m = 0; m < 16; ++m) {
        const int r = row0 + m;
        const float x0 = x[r * 3 + 0];
        const float x1 = x[r * 3 + 1];
        const float x2 = x[r * 3 + 2];
#pragma unroll
        for (int jj = 0; jj < 4; ++jj) {
            const int j = jj * 32 + lane;
            const float u = x0 * B[0 * 128 + j] + x1 * B[1 * 128 + j] + x2 * B[2 * 128 + j];
            ffm[m * K_PROJ + j]       = __builtin_amdgcn_cosf(u);
            ffm[m * K_PROJ + 128 + j] = __builtin_amdgcn_sinf(u);
        }
    }
    __syncthreads();

    const int col  = lane & 15;
    const int koff = (lane >> 4) * 2;   // lanes 0-15: K+0/K+1, lanes 16-31: K+2/K+3
    const int arow = lane & 15;

    v8f acc[6];
#pragma unroll
    for (int t = 0; t < 6; ++t) {
        const float bv = bp[t * 16 + col];
        v8f ini = {bv, bv, bv, bv, bv, bv, bv, bv};
        acc[t] = ini;
    }

    for (int kb = 0; kb < K_PROJ / 4; ++kb) {
        const int k = kb * 4 + koff;
        v2f a;
        a.x = ffm[arow * K_PROJ + k];
        a.y = ffm[arow * K_PROJ + k + 1];
#pragma unroll
        for (int t = 0; t < 6; ++t) {
            v2f b;
            b.x = Wp[k * F_OUT + t * 16 + col];
            b.y = Wp[(k + 1) * F_OUT + t * 16 + col];
            acc[t] = __builtin_amdgcn_wmma_f32_16x16x4_f32(
                false, a, false, b, (short)0, acc[t], false, false);
        }
    }

#pragma unroll
    for (int t = 0; t < 6; ++t) {
#pragma unroll
        for (int v = 0; v < 8; ++v) {
            const int m = v + ((lane >> 4) << 3);       // C/D layout: lanes>=16 hold M+8
            x_embed[(size_t)(row0 + m) * F_OUT + t * 16 + col] = acc[t][v];
        }
    }
}

// ---------------------------------------------------------------------------
// Kernel 2: edge scatter. One wave per edge; 96 features = 3 per lane.
// Everything (src rows, dst rows) is L2-resident; native f32 atomics.
// ---------------------------------------------------------------------------
__global__ __launch_bounds__(256) void scatter_kernel(
    const long long* __restrict__ ei, const float* __restrict__ x_embed,
    float* __restrict__ agg, float* __restrict__ deg, int nedges)
{
    const int e    = (int)((blockIdx.x * (unsigned)blockDim.x + threadIdx.x) >> 5);
    const int lane = threadIdx.x & 31;
    if (e >= nedges) return;

    const long long s = ei[e];             // edge_index[0][e] (src)
    const long long d = ei[nedges + e];    // edge_index[1][e] (dst)

    const float* __restrict__ srcv = x_embed + (size_t)s * F_OUT;
    float* __restrict__ dstv = agg + (size_t)d * F_OUT;

    unsafeAtomicAdd(dstv + lane,      srcv[lane]);
    unsafeAtomicAdd(dstv + lane + 32, srcv[lane + 32]);
    unsafeAtomicAdd(dstv + lane + 64, srcv[lane + 64]);
    if (lane == 0) unsafeAtomicAdd(deg + d, 1.0f);
}

// ---------------------------------------------------------------------------
// Kernel 3: fusion GEMM. combined = [x_embed | agg/max(deg,1)] (16x192 in LDS),
// fp32 WMMA against W_fus, + bias. Writes fused into d_out and accumulates
// per-column sum / sum-of-squares for BatchNorm batch stats.
// ---------------------------------------------------------------------------
__global__ __launch_bounds__(32) void fuse_kernel(
    const float* __restrict__ x_embed, const float* __restrict__ agg,
    const float* __restrict__ deg, const float* __restrict__ Wf,
    const float* __restrict__ bf, float* __restrict__ fused,
    float* __restrict__ colsum, float* __restrict__ colsumsq)
{
    __shared__ float comb[16 * K_FUS];
    const int lane = threadIdx.x;
    const int row0 = blockIdx.x * 16;

    for (int m = 0; m < 16; ++m) {
        const int r = row0 + m;
        const float invd = 1.0f / fmaxf(deg[r], 1.0f);
#pragma unroll
        for (int jj = 0; jj < 3; ++jj) {
            const int c = jj * 32 + lane;
            comb[m * K_FUS + c]      = x_embed[(size_t)r * F_OUT + c];
            comb[m * K_FUS + 96 + c] = agg[(size_t)r * F_OUT + c] * invd;
        }
    }
    __syncthreads();

    const int col  = lane & 15;
    const int koff = (lane >> 4) * 2;
    const int arow = lane & 15;

    v8f acc[6];
#pragma unroll
    for (int t = 0; t < 6; ++t) {
        const float bv = bf[t * 16 + col];
        v8f ini = {bv, bv, bv, bv, bv, bv, bv, bv};
        acc[t] = ini;
    }

    for (int kb = 0; kb < K_FUS / 4; ++kb) {
        const int k = kb * 4 + koff;
        v2f a;
        a.x = comb[arow * K_FUS + k];
        a.y = comb[arow * K_FUS + k + 1];
#pragma unroll
        for (int t = 0; t < 6; ++t) {
            v2f b;
            b.x = Wf[k * F_OUT + t * 16 + col];
            b.y = Wf[(k + 1) * F_OUT + t * 16 + col];
            acc[t] = __builtin_amdgcn_wmma_f32_16x16x4_f32(
                false, a, false, b, (short)0, acc[t], false, false);
        }
    }

#pragma unroll
    for (int t = 0; t < 6; ++t) {
        float s = 0.0f, ss = 0.0f;
#pragma unroll
        for (int v = 0; v < 8; ++v) {
            const int m = v + ((lane >> 4) << 3);
            const float f = acc[t][v];
            fused[(size_t)(row0 + m) * F_OUT + t * 16 + col] = f;
            s  += f;
            ss += f * f;
        }
        unsafeAtomicAdd(colsum   + t * 16 + col, s);
        unsafeAtomicAdd(colsumsq + t * 16 + col, ss);
    }
}

// ---------------------------------------------------------------------------
// Kernel 4: finalize batch stats -> per-channel affine (scale, shift).
// ---------------------------------------------------------------------------
__global__ __launch_bounds__(128) void stats_kernel(
    const float* __restrict__ colsum, const float* __restrict__ colsumsq,
    const float* __restrict__ gamma, const float* __restrict__ beta,
    float* __restrict__ scale, float* __restrict__ shift)
{
    const int c = threadIdx.x;
    if (c < F_OUT) {
        const float invN = 1.0f / (float)N_NODES;
        const float mean = colsum[c] * invN;
        const float var  = colsumsq[c] * invN - mean * mean;
        const float sc   = rsqrtf(var + BN_EPS) * gamma[c];
        scale[c] = sc;
        shift[c] = beta[c] - mean * sc;
    }
}

// ---------------------------------------------------------------------------
// Kernel 5: BN affine + exact GELU, in place on d_out.
// ---------------------------------------------------------------------------
__global__ __launch_bounds__(256) void bn_gelu_kernel(
    float* __restrict__ out, const float* __restrict__ scale,
    const float* __restrict__ shift, int total)
{
    const int i = blockIdx.x * blockDim.x + threadIdx.x;
    if (i < total) {
        const int c = i % F_OUT;
        const float y = out[i] * scale[c] + shift[c];
        out[i] = 0.5f * y * (1.0f + erff(y * 0.70710678118654752f));
    }
}

// ---------------------------------------------------------------------------
extern "C" void kernel_launch(void* const* d_in, const int* in_sizes, int n_in,
                              void* d_out, int out_size, void* d_ws, size_t ws_size,
                              hipStream_t stream) {
    const float*      x     = (const float*)d_in[0];
    const float*      B     = (const float*)d_in[1];
    const float*      Wp    = (const float*)d_in[2];
    const float*      bp    = (const float*)d_in[3];
    const float*      Wf    = (const float*)d_in[4];
    const float*      bf    = (const float*)d_in[5];
    const float*      gamma = (const float*)d_in[6];
    const float*      beta  = (const float*)d_in[7];
    const long long*  ei    = (const long long*)d_in[8];   // int64 [2, E]
    const int E = in_sizes[8] / 2;

    // workspace layout (floats)
    float* ws       = (float*)d_ws;
    float* x_embed  = ws;                                  // 4,800,000
    float* agg      = ws + 4800000;                        // 4,800,000
    float* deg      = ws + 9600000;                        //    50,000
    float* colsum   = ws + 9650000;                        //        96
    float* colsumsq = ws + 9650096;                        //        96
    float* scale    = ws + 9650192;                        //        96
    float* shift    = ws + 9650288;                        //        96

    // zero agg + deg + colsum + colsumsq (contiguous) — required every call
    hipMemsetAsync(agg, 0, (size_t)(4800000 + 50000 + 192) * sizeof(float), stream);

    ffm_proj_kernel<<<NTILES, 32, 0, stream>>>(x, B, Wp, bp, x_embed);

    const int scatter_threads = E * 32;
    scatter_kernel<<<(scatter_threads + 255) / 256, 256, 0, stream>>>(
        ei, x_embed, agg, deg, E);

    fuse_kernel<<<NTILES, 32, 0, stream>>>(
        x_embed, agg, deg, Wf, bf, (float*)d_out, colsum, colsumsq);

    stats_kernel<<<1, 128, 0, stream>>>(colsum, colsumsq, gamma, beta, scale, shift);

    const int total = N_NODES * F_OUT;
    bn_gelu_kernel<<<(total + 255) / 256, 256, 0, stream>>>(
        (float*)d_out, scale, shift, total);
}